// PrototypeTripletRankingLoss_78632261255707
// MI455X (gfx1250) — compile-verified
//
#include <hip/hip_runtime.h>
#include <hip/hip_bf16.h>

// Triplet cosine-similarity loss, B=1024, K=127, D=512, f32.
// Memory-bound (AI ~1 FLOP/byte): optimize data movement, not matrix math.
// CDNA5 paths used: global_load_async_to_lds_b128 (+ s_wait_asynccnt) to stage
// the reused ref vector in LDS, non-temporal b128 streaming loads for neg,
// global_prefetch_b8 lookahead. wave32 shuffles throughout.

#define TL_B 1024
#define TL_K 127
#define TL_D 512
#define TL_MARGIN 0.2f
#define TL_EPS 1e-8f

typedef float f32x4 __attribute__((ext_vector_type(4)));
typedef float f32x2 __attribute__((ext_vector_type(2)));

__global__ __launch_bounds__(256) void triplet_main_kernel(
    const float* __restrict__ ref, const float* __restrict__ pos,
    const float* __restrict__ neg, float* __restrict__ partial) {
  __shared__ __align__(16) float s_ref[TL_D];  // 2KB, reused 127x per block
  __shared__ float s_red[24];                  // 3 values x 8 waves
  __shared__ float s_scal[2];                  // rnorm, pos_sim
  __shared__ float s_wsum[8];

  const int b    = blockIdx.x;
  const int tid  = threadIdx.x;
  const int lane = tid & 31;
  const int wave = tid >> 5;

  const float* refb = ref + (size_t)b * TL_D;
  const float* posb = pos + (size_t)b * TL_D;

  // ---- Stage ref[b] into LDS via CDNA5 async global->LDS DMA path ----
  // 512 floats = 128 x b128 transfers, issued by lanes of waves 0..3.
  if (tid < 128) {
    unsigned lds_off = (unsigned)(uintptr_t)(&s_ref[0]) + (unsigned)tid * 16u;
    const float* g = refb + tid * 4;
    asm volatile("global_load_async_to_lds_b128 %0, %1, off"
                 :: "v"(lds_off), "v"(g)
                 : "memory");
  }
  asm volatile("s_wait_asynccnt 0x0" ::: "memory");

  // pos can be read while the DMA is in flight
  f32x2 p = ((const f32x2*)posb)[tid];  // 2 floats/thread * 256 = 512

  __syncthreads();  // make async-LDS writes visible to all waves

  f32x2 r = ((const f32x2*)s_ref)[tid];

  float rr = r.x * r.x + r.y * r.y;
  float pp = p.x * p.x + p.y * p.y;
  float rp = r.x * p.x + r.y * p.y;

  // wave32 butterfly reductions (fixed shape -> deterministic)
  #pragma unroll
  for (int off = 16; off > 0; off >>= 1) {
    rr += __shfl_xor(rr, off, 32);
    pp += __shfl_xor(pp, off, 32);
    rp += __shfl_xor(rp, off, 32);
  }
  if (lane == 0) {
    s_red[wave]      = rr;
    s_red[8 + wave]  = rp;
    s_red[16 + wave] = pp;
  }
  __syncthreads();
  if (tid == 0) {
    float a = 0.f, c = 0.f, d = 0.f;
    #pragma unroll
    for (int i = 0; i < 8; ++i) { a += s_red[i]; c += s_red[8 + i]; d += s_red[16 + i]; }
    float rnorm = fmaxf(sqrtf(a), TL_EPS);
    float pnorm = fmaxf(sqrtf(d), TL_EPS);
    s_scal[0] = rnorm;
    s_scal[1] = c / (rnorm * pnorm);  // pos_sim
  }
  __syncthreads();
  const float rnorm  = s_scal[0];
  const float possim = s_scal[1];

  // ---- Stream neg[b,k,:] : each wave owns k = wave, wave+8, ... ----
  const float* negb = neg + (size_t)b * TL_K * TL_D;
  float wacc = 0.f;
  for (int k = wave; k < TL_K; k += 8) {
    const float* nk = negb + (size_t)k * TL_D;

    // lookahead: one prefetch op covers the whole next 2KB row
    // (lane*64B stride -> 32 lanes span 2048B)
    if (k + 8 < TL_K) {
      __builtin_prefetch(negb + (size_t)(k + 8) * TL_D + lane * 16, 0, 0);
    }

    float dot = 0.f, nn = 0.f;
    #pragma unroll
    for (int i = 0; i < 4; ++i) {
      const int idx = lane * 4 + i * 128;           // coalesced 512B/instr
      f32x4 nv = __builtin_nontemporal_load((const f32x4*)(nk + idx));
      f32x4 rv = *(const f32x4*)(s_ref + idx);      // ds_load_b128
      dot = fmaf(rv.x, nv.x, dot);
      dot = fmaf(rv.y, nv.y, dot);
      dot = fmaf(rv.z, nv.z, dot);
      dot = fmaf(rv.w, nv.w, dot);
      nn  = fmaf(nv.x, nv.x, nn);
      nn  = fmaf(nv.y, nv.y, nn);
      nn  = fmaf(nv.z, nv.z, nn);
      nn  = fmaf(nv.w, nv.w, nn);
    }
    #pragma unroll
    for (int off = 16; off > 0; off >>= 1) {
      dot += __shfl_xor(dot, off, 32);
      nn  += __shfl_xor(nn,  off, 32);
    }
    float nnorm = fmaxf(sqrtf(nn), TL_EPS);
    float sim   = dot / (rnorm * nnorm);
    wacc += fmaxf(TL_MARGIN + sim - possim, 0.f);
  }

  if (lane == 0) s_wsum[wave] = wacc;
  __syncthreads();
  if (tid == 0) {
    float s = 0.f;
    #pragma unroll
    for (int i = 0; i < 8; ++i) s += s_wsum[i];
    partial[b] = s * (1.0f / (float)TL_K);  // mean over K
  }
}

// Deterministic fixed-order reduction of 1024 partials -> scalar mean.
__global__ __launch_bounds__(256) void triplet_reduce_kernel(
    const float* __restrict__ partial, float* __restrict__ out) {
  __shared__ float s[8];
  const int tid  = threadIdx.x;
  const int lane = tid & 31;
  const int wave = tid >> 5;

  float v = partial[tid] + partial[tid + 256] +
            partial[tid + 512] + partial[tid + 768];
  #pragma unroll
  for (int off = 16; off > 0; off >>= 1) v += __shfl_xor(v, off, 32);
  if (lane == 0) s[wave] = v;
  __syncthreads();
  if (tid == 0) {
    float t = 0.f;
    #pragma unroll
    for (int i = 0; i < 8; ++i) t += s[i];
    out[0] = t * (1.0f / (float)TL_B);  // mean over B
  }
}

extern "C" void kernel_launch(void* const* d_in, const int* in_sizes, int n_in,
                              void* d_out, int out_size, void* d_ws, size_t ws_size,
                              hipStream_t stream) {
  (void)in_sizes; (void)n_in; (void)out_size; (void)ws_size;
  const float* ref = (const float*)d_in[0];
  const float* pos = (const float*)d_in[1];
  const float* neg = (const float*)d_in[2];
  float* out     = (float*)d_out;
  float* partial = (float*)d_ws;  // 1024 floats = 4KB scratch

  triplet_main_kernel<<<TL_B, 256, 0, stream>>>(ref, pos, neg, partial);
  triplet_reduce_kernel<<<1, 256, 0, stream>>>(partial, out);
}